// BiBoxLoss_79594333929975
// MI455X (gfx1250) — compile-verified
//
#include <hip/hip_runtime.h>
#include <stdint.h>

typedef __attribute__((ext_vector_type(2))) float v2f;
typedef __attribute__((ext_vector_type(8))) float v8f;

#define BATCH   32
#define PRIORS  65536
#define NTRUTH  16
#define THRESH  0.5f

// ---------------------------------------------------------------------------
// Wave(32)-wide sum using V_WMMA_F32_16X16X4_F32 (ones-matrix reduction):
// A (16x4): vgpr0 = per-lane value x  => A[M,0]=x_M (lanes 0-15), A[M,2]=x_{M+16}
// B (4x16): all ones. D[M,c] = x_M + x_{M+16}. Per lane, D components j=0..7
// hold 8 row sums; adding them + the xor-16 partner gives the full 32-lane sum.
// Requires EXEC == all ones (callers guarantee full blocks).
// ---------------------------------------------------------------------------
__device__ __forceinline__ float wave_sum_wmma(float x) {
  v2f a; a[0] = x;    a[1] = 0.0f;
  v2f b; b[0] = 1.0f; b[1] = 1.0f;
  v8f c = {0.f, 0.f, 0.f, 0.f, 0.f, 0.f, 0.f, 0.f};
  c = __builtin_amdgcn_wmma_f32_16x16x4_f32(false, a, false, b, (short)0, c,
                                            false, false);
  float s = 0.0f;
#pragma unroll
  for (int i = 0; i < 8; ++i) s += c[i];
  s += __shfl_xor(s, 16, 32);
  return s;
}

__device__ __forceinline__ float smooth_l1(float d) {
  float a = fabsf(d);
  return (a < 1.0f) ? 0.5f * a * a : a - 0.5f;
}

// ---------------------------------------------------------------------------
// K0: zero small accumulators (big ws arrays are fully overwritten each call)
// ---------------------------------------------------------------------------
__global__ void k_init(unsigned long long* tbest, int* npos, float* accs) {
  int i = blockIdx.x * 256 + threadIdx.x;
  if (i < BATCH * NTRUTH) tbest[i] = 0ull;
  if (i < BATCH) npos[i] = 0;
  if (i < 2) accs[i] = 0.0f;
}

// ---------------------------------------------------------------------------
// K1: per-prior best truth (max/argmax over 16 truths) + per-truth best prior
// (argmax over priors via packed atomicMax). Truths staged into LDS with the
// CDNA5 async global->LDS path.
// ---------------------------------------------------------------------------
__global__ void k_match(const float* __restrict__ priors,
                        const float* __restrict__ targets,
                        unsigned long long* __restrict__ packed,
                        unsigned long long* __restrict__ tbest) {
  __shared__ float tr[NTRUTH * 4];
  __shared__ unsigned long long tb[NTRUTH];
  const int tid = threadIdx.x;
  const int b = blockIdx.y;
  const int p = blockIdx.x * 256 + tid;

  if (tid < NTRUTH) tb[tid] = 0ull;

  // Async copy: 16 truths x 4 coords (targets row stride is 5 floats).
  if (tid < 64) {
    unsigned ldsOff = (unsigned)(uintptr_t)(&tr[tid]);  // low 32 bits == LDS offset
    const float* gsrc =
        targets + (size_t)b * NTRUTH * 5 + (size_t)(tid >> 2) * 5 + (tid & 3);
    unsigned long long gaddr = (unsigned long long)(uintptr_t)gsrc;
    asm volatile("global_load_async_to_lds_b32 %0, %1, off"
                 :: "v"(ldsOff), "v"(gaddr) : "memory");
  }
  asm volatile("s_wait_asynccnt 0" ::: "memory");
  __syncthreads();

  const float4 pr = ((const float4*)priors)[p];
  const float plx = pr.x - pr.z * 0.5f, ply = pr.y - pr.w * 0.5f;
  const float prx = pr.x + pr.z * 0.5f, pry = pr.y + pr.w * 0.5f;
  const float area_p = (prx - plx) * (pry - ply);

  float best = -1.0f;
  int bidx = 0;
#pragma unroll
  for (int t = 0; t < NTRUTH; ++t) {
    const float t0 = tr[t * 4 + 0], t1 = tr[t * 4 + 1];
    const float t2 = tr[t * 4 + 2], t3 = tr[t * 4 + 3];
    const float ltx = fmaxf(t0, plx), lty = fmaxf(t1, ply);
    const float rbx = fminf(t2, prx), rby = fminf(t3, pry);
    const float w = fmaxf(rbx - ltx, 0.0f), h = fmaxf(rby - lty, 0.0f);
    const float inter = w * h;
    const float area_t = (t2 - t0) * (t3 - t1);
    const float iou = inter / (area_t + area_p - inter);
    if (iou > best) { best = iou; bidx = t; }
    // per-truth argmax over priors: value bits high, ~p low (tie -> smaller p)
    const unsigned long long pkt =
        ((unsigned long long)__float_as_uint(iou) << 32) | (unsigned)(~(unsigned)p);
    atomicMax(&tb[t], pkt);
  }
  packed[(size_t)b * PRIORS + p] =
      ((unsigned long long)__float_as_uint(best) << 32) | (unsigned)bidx;

  __syncthreads();
  if (tid < NTRUTH) atomicMax(&tbest[b * NTRUTH + tid], tb[tid]);
}

// ---------------------------------------------------------------------------
// K2: scatter override: best_truth_overlap[best_prior_idx[t]] = 2.0, idx = t.
// Sequential t per batch so duplicates resolve "last t wins".
// ---------------------------------------------------------------------------
__global__ void k_override(const unsigned long long* __restrict__ tbest,
                           unsigned long long* __restrict__ packed) {
  const int b = threadIdx.x;
  if (b >= BATCH) return;
  for (int t = 0; t < NTRUTH; ++t) {
    const unsigned long long pk = tbest[b * NTRUTH + t];
    const unsigned p = ~(unsigned)(pk & 0xFFFFFFFFull);
    if (p < PRIORS) {
      packed[(size_t)b * PRIORS + p] =
          ((unsigned long long)__float_as_uint(2.0f) << 32) | (unsigned)t;
    }
  }
}

// ---------------------------------------------------------------------------
// K3: streaming loss pass: encode, smooth-L1 (masked), pos count, logsumexp.
// ---------------------------------------------------------------------------
__global__ void k_loss(const float* __restrict__ loc,
                       const float* __restrict__ conf,
                       const float* __restrict__ priors,
                       const float* __restrict__ targets,
                       const unsigned long long* __restrict__ packed,
                       float* __restrict__ lca, int* __restrict__ npos,
                       float* __restrict__ accs) {
  __shared__ float tr[NTRUTH * 4];
  const int tid = threadIdx.x;
  const int b = blockIdx.y;
  const int p = blockIdx.x * 256 + tid;
  if (tid < 64)
    tr[tid] = targets[(size_t)b * NTRUTH * 5 + (size_t)(tid >> 2) * 5 + (tid & 3)];
  __syncthreads();

  const unsigned long long pk = packed[(size_t)b * PRIORS + p];
  const float ov = __uint_as_float((unsigned)(pk >> 32));
  const int ti = (int)(pk & 15ull);
  const bool pos = (ov >= THRESH);

  const float4 pr = ((const float4*)priors)[p];
  const float m0 = tr[ti * 4 + 0], m1 = tr[ti * 4 + 1];
  const float m2 = tr[ti * 4 + 2], m3 = tr[ti * 4 + 3];
  const float gx = ((m0 + m2) * 0.5f - pr.x) / (0.1f * pr.z);
  const float gy = ((m1 + m3) * 0.5f - pr.y) / (0.1f * pr.w);
  const float gw = logf((m2 - m0) / pr.z) * 5.0f;  // /VARIANCE[1]
  const float gh = logf((m3 - m1) / pr.w) * 5.0f;

  const float4 l = ((const float4*)loc)[(size_t)b * PRIORS + p];
  float s = smooth_l1(l.x - gx) + smooth_l1(l.y - gy) +
            smooth_l1(l.z - gw) + smooth_l1(l.w - gh);
  s = pos ? s : 0.0f;

  const float ws = wave_sum_wmma(s);  // full blocks -> EXEC all ones
  const unsigned bal = (unsigned)__ballot(pos);
  if ((tid & 31) == 0) {
    atomicAdd(&accs[0], ws);
    atomicAdd(&npos[b], __popc(bal));
  }

  const float2 c = ((const float2*)conf)[(size_t)b * PRIORS + p];
  const float mx = fmaxf(c.x, c.y);
  const float lse = mx + logf(expf(c.x - mx) + expf(c.y - mx));
  const float g = pos ? c.y : c.x;
  lca[(size_t)b * PRIORS + p] = lse - g;
}

// ---------------------------------------------------------------------------
// K4: hard-negative mining per batch row. Exact radix-select (4x8-bit passes
// over float bit patterns; values >= 0 so uint order == float order), then an
// index-ordered tie-resolving selection matching stable argsort ranking.
// ---------------------------------------------------------------------------
__global__ void k_mine(const unsigned long long* __restrict__ packed,
                       const float* __restrict__ lca,
                       const int* __restrict__ npos,
                       float* __restrict__ accs) {
  __shared__ unsigned hist[256];
  __shared__ unsigned sh_prefix, sh_rem;
  __shared__ unsigned wcnt[8];
  __shared__ unsigned tie_base;
  __shared__ float wsum[8];

  const int b = blockIdx.x;
  const int tid = threadIdx.x;
  const int lane = tid & 31, wv = tid >> 5;
  const unsigned long long* pk = packed + (size_t)b * PRIORS;
  const float* lc = lca + (size_t)b * PRIORS;

  const int np = npos[b];
  const int K = min(3 * np, PRIORS - 1);

  unsigned prefix = 0, pmask = 0, remaining = (unsigned)K;
  unsigned thr, need;
  if (K > 0) {
    for (int pass = 0; pass < 4; ++pass) {
      const int shift = 24 - 8 * pass;
      hist[tid] = 0;
      __syncthreads();
      for (int it = 0; it < PRIORS / 256; ++it) {
        const int i = it * 256 + tid;
        if (it + 1 < PRIORS / 256) __builtin_prefetch(&lc[i + 256], 0, 3);
        const unsigned long long q = pk[i];
        const bool pos = __uint_as_float((unsigned)(q >> 32)) >= THRESH;
        const unsigned u = pos ? 0u : __float_as_uint(lc[i]);
        if ((u & pmask) == prefix) atomicAdd(&hist[(u >> shift) & 255], 1u);
      }
      __syncthreads();
      if (tid == 0) {
        unsigned cum = 0;
        for (int d = 255; d >= 0; --d) {
          const unsigned cc = hist[d];
          if (cum + cc >= remaining) {
            sh_prefix = prefix | ((unsigned)d << shift);
            sh_rem = remaining - cum;  // ties still needed within this bin
            break;
          }
          cum += cc;
        }
      }
      __syncthreads();
      prefix = sh_prefix;
      remaining = sh_rem;
      pmask |= (0xFFu << shift);
      __syncthreads();
    }
    thr = prefix;
    need = remaining;
  } else {
    thr = 0xFFFFFFFFu;
    need = 0u;
  }

  if (tid == 0) tie_base = 0;
  __syncthreads();

  float acc = 0.0f;
  for (int it = 0; it < PRIORS / 256; ++it) {
    const int i = it * 256 + tid;
    const unsigned long long q = pk[i];
    const bool pos = __uint_as_float((unsigned)(q >> 32)) >= THRESH;
    const float v = lc[i];
    const unsigned u = pos ? 0u : __float_as_uint(v);
    const bool gt = (u > thr);
    const bool tie = (u == thr);
    const unsigned mask = (unsigned)__ballot(tie);
    if (lane == 0) wcnt[wv] = (unsigned)__popc(mask);
    __syncthreads();
    unsigned off = tie_base;
    for (int j = 0; j < wv; ++j) off += wcnt[j];
    const unsigned rank = off + (unsigned)__popc(mask & ((1u << lane) - 1u));
    const bool seltie = tie && (rank < need);
    if (pos || gt || seltie) acc += v;
    __syncthreads();
    if (tid == 0) {
      unsigned tot = 0;
      for (int j = 0; j < 8; ++j) tot += wcnt[j];
      tie_base += tot;
    }
    __syncthreads();
  }

  const float wsv = wave_sum_wmma(acc);  // full block -> EXEC all ones
  if (lane == 0) wsum[wv] = wsv;
  __syncthreads();
  if (tid == 0) {
    float t = 0.0f;
    for (int j = 0; j < 8; ++j) t += wsum[j];
    atomicAdd(&accs[1], t);
  }
}

// ---------------------------------------------------------------------------
// K5: finalize
// ---------------------------------------------------------------------------
__global__ void k_final(const int* __restrict__ npos,
                        const float* __restrict__ accs,
                        float* __restrict__ out) {
  if (threadIdx.x == 0 && blockIdx.x == 0) {
    int n = 0;
    for (int b = 0; b < BATCH; ++b) n += npos[b];
    const float N = (float)n;
    out[0] = accs[0] / N;
    out[1] = accs[1] / N;
  }
}

// ---------------------------------------------------------------------------
extern "C" void kernel_launch(void* const* d_in, const int* in_sizes, int n_in,
                              void* d_out, int out_size, void* d_ws,
                              size_t ws_size, hipStream_t stream) {
  (void)in_sizes; (void)n_in; (void)out_size; (void)ws_size;
  const float* loc     = (const float*)d_in[0];  // (32, 65536, 4)
  const float* conf    = (const float*)d_in[1];  // (32, 65536, 2)
  const float* priors  = (const float*)d_in[2];  // (65536, 4)
  const float* targets = (const float*)d_in[3];  // (32, 16, 5)
  float* out = (float*)d_out;                    // 2 floats

  char* ws = (char*)d_ws;
  const size_t OFF_PACKED = 0;
  const size_t OFF_LCA    = OFF_PACKED + (size_t)BATCH * PRIORS * 8;  // 16 MB
  const size_t OFF_TBEST  = OFF_LCA + (size_t)BATCH * PRIORS * 4;     // +8 MB
  const size_t OFF_NPOS   = OFF_TBEST + (size_t)BATCH * NTRUTH * 8;   // +4 KB
  const size_t OFF_ACC    = OFF_NPOS + 128;

  unsigned long long* packed = (unsigned long long*)(ws + OFF_PACKED);
  float* lca                 = (float*)(ws + OFF_LCA);
  unsigned long long* tbest  = (unsigned long long*)(ws + OFF_TBEST);
  int* npos                  = (int*)(ws + OFF_NPOS);
  float* accs                = (float*)(ws + OFF_ACC);

  k_init<<<2, 256, 0, stream>>>(tbest, npos, accs);
  k_match<<<dim3(PRIORS / 256, BATCH), 256, 0, stream>>>(priors, targets,
                                                         packed, tbest);
  k_override<<<1, 32, 0, stream>>>(tbest, packed);
  k_loss<<<dim3(PRIORS / 256, BATCH), 256, 0, stream>>>(
      loc, conf, priors, targets, packed, lca, npos, accs);
  k_mine<<<BATCH, 256, 0, stream>>>(packed, lca, npos, accs);
  k_final<<<1, 1, 0, stream>>>(npos, accs, out);
}